// SimCLR_47253230191428
// MI455X (gfx1250) — compile-verified
//
#include <hip/hip_runtime.h>
#include <math.h>

#define D_DIM 128
#define COL_SPLIT 8
#define PAD_ROW 132   // 128 + 4 floats: conflict-free ds_load_b64 fragment reads
// rows scaled by sqrt(1/T) so WMMA output is already sim/T, bounded in [-2,2]
#define ROW_SCALE 1.41421356237309515f

typedef __attribute__((ext_vector_type(2))) float v2f;
typedef __attribute__((ext_vector_type(8))) float v8f;

__device__ __forceinline__ float wave_allreduce_add(float x) {
#pragma unroll
  for (int m = 16; m >= 1; m >>= 1) x += __shfl_xor(x, m, 32);
  return x;
}

// Kernel 1: normalize z1/z2 rows, scale by sqrt(1/T), store concatenated Z,
// and compute pos[i] = <z1n_i, z2n_i>/T.  One wave32 per row pair (4 f32/lane).
__global__ void __launch_bounds__(256) simclr_norm(const float* __restrict__ z1,
                                                   const float* __restrict__ z2,
                                                   float* __restrict__ zs,
                                                   float* __restrict__ pos, int B) {
  const int lane = threadIdx.x & 31;
  const int wave = threadIdx.x >> 5;
  const int row  = blockIdx.x * 8 + wave;
  if (row >= B) return;
  const float4 a = ((const float4*)(z1 + (size_t)row * D_DIM))[lane];
  const float4 b = ((const float4*)(z2 + (size_t)row * D_DIM))[lane];
  float ss1 = a.x * a.x + a.y * a.y + a.z * a.z + a.w * a.w;
  float ss2 = b.x * b.x + b.y * b.y + b.z * b.z + b.w * b.w;
  ss1 = wave_allreduce_add(ss1);
  ss2 = wave_allreduce_add(ss2);
  const float inv1 = ROW_SCALE / fmaxf(sqrtf(ss1), 1e-12f);
  const float inv2 = ROW_SCALE / fmaxf(sqrtf(ss2), 1e-12f);
  const float4 as = make_float4(a.x * inv1, a.y * inv1, a.z * inv1, a.w * inv1);
  const float4 bs = make_float4(b.x * inv2, b.y * inv2, b.z * inv2, b.w * inv2);
  ((float4*)(zs + (size_t)row * D_DIM))[lane]       = as;
  ((float4*)(zs + (size_t)(B + row) * D_DIM))[lane] = bs;
  float p = as.x * bs.x + as.y * bs.y + as.z * bs.z + as.w * bs.w;
  p = wave_allreduce_add(p);
  if (lane == 0) pos[row] = p;  // already divided by T (scaled vectors)
}

// Kernel 2: block = 8 row tiles x 1 column chunk.  B tiles (16x128 f32, 8KB)
// are double-buffered in LDS and shared by all 8 waves; global loads for the
// next tile are in flight during the current tile's 32 WMMAs + 8 exps.
// Diagonal is NOT masked: it contributes exactly exp(0)=1, removed later.
__global__ void __launch_bounds__(256) simclr_sim(const float* __restrict__ zs,
                                                  float* __restrict__ partial,
                                                  int N, int colTiles) {
  __shared__ __align__(16) float lbuf[2][16 * PAD_ROW];

  const int lane     = threadIdx.x & 31;
  const int wave     = threadIdx.x >> 5;
  const int chunk    = blockIdx.x & (COL_SPLIT - 1);
  const int rowGroup = blockIdx.x / COL_SPLIT;
  const int rowbase  = (rowGroup * 8 + wave) * 16;
  const int m  = lane & 15;           // M for A-frag, N for B/C-frag
  const int kb = (lane >> 4) * 2;     // K phase within each 4-wide chunk

  // cooperative staging map: each thread copies 8 floats of one B-tile row
  const int srow = threadIdx.x >> 4;        // 0..15
  const int scol = (threadIdx.x & 15) * 8;  // 0,8,...,120

  // A fragments: registers for the whole sweep (64 VGPRs)
  const float* __restrict__ arow = zs + (size_t)(rowbase + m) * D_DIM + kb;
  v2f a[32];
#pragma unroll
  for (int kc = 0; kc < 32; ++kc) a[kc] = *(const v2f*)(arow + kc * 4);

  float s[8];
#pragma unroll
  for (int v = 0; v < 8; ++v) s[v] = 0.0f;

  const int colbase0 = chunk * (colTiles * 16);

  // stage tile 0
  {
    const float* src = zs + (size_t)(colbase0 + srow) * D_DIM + scol;
    const float4 t0 = ((const float4*)src)[0];
    const float4 t1 = ((const float4*)src)[1];
    *(float4*)&lbuf[0][srow * PAD_ROW + scol]     = t0;
    *(float4*)&lbuf[0][srow * PAD_ROW + scol + 4] = t1;
  }
  __syncthreads();

  for (int ct = 0; ct < colTiles; ++ct) {
    const int cur = ct & 1;
    const bool more = (ct + 1) < colTiles;  // wave-uniform -> scalar branch
    float4 t0 = make_float4(0.f, 0.f, 0.f, 0.f);
    float4 t1 = make_float4(0.f, 0.f, 0.f, 0.f);
    if (more) {  // issue next tile's global loads; they fly during compute
      const float* src =
          zs + (size_t)(colbase0 + (ct + 1) * 16 + srow) * D_DIM + scol;
      t0 = ((const float4*)src)[0];
      t1 = ((const float4*)src)[1];
    }

    // compute 16x16 sim tile from LDS (conflict-free ds_load_b64 fragments)
    const float* bfrag = &lbuf[cur][m * PAD_ROW + kb];
    v8f c = {0.f, 0.f, 0.f, 0.f, 0.f, 0.f, 0.f, 0.f};
#pragma unroll
    for (int kc = 0; kc < 32; ++kc) {
      const v2f b = *(const v2f*)(bfrag + kc * 4);
      c = __builtin_amdgcn_wmma_f32_16x16x4_f32(false, a[kc], false, b,
                                                (short)0, c, false, false);
    }
#pragma unroll
    for (int v = 0; v < 8; ++v) s[v] += __expf(c[v] - 2.0f);

    if (more) {
      float* dst = &lbuf[1 - cur][srow * PAD_ROW + scol];
      *(float4*)dst       = t0;
      *(float4*)(dst + 4) = t1;
    }
    __syncthreads();
  }

  // Sum over the 16 lanes (same half) that cover this row set's columns.
#pragma unroll
  for (int v = 0; v < 8; ++v) {
    float x = s[v];
#pragma unroll
    for (int msk = 8; msk >= 1; msk >>= 1) x += __shfl_xor(x, msk, 32);
    s[v] = x;
  }
  if (m == 0) {  // lanes 0 (rows 0..7) and 16 (rows 8..15)
    const int mb = (lane >> 4) * 8;
#pragma unroll
    for (int v = 0; v < 8; ++v)
      partial[(size_t)chunk * N + rowbase + mb + v] = s[v];
  }
}

// Kernel 3: deterministic reduction.  lse_r = 2 + log(sum_chunks - 1),
// loss = mean(lse - pos).
__global__ void __launch_bounds__(256) simclr_final(const float* __restrict__ partial,
                                                    const float* __restrict__ pos,
                                                    float* __restrict__ out,
                                                    int N, int B) {
  __shared__ float red[256];
  float acc = 0.0f;
  for (int r = threadIdx.x; r < N; r += 256) {
    float s = 0.0f;
#pragma unroll
    for (int c = 0; c < COL_SPLIT; ++c) s += partial[(size_t)c * N + r];
    s -= 1.0f;  // remove self-similarity term exp(2-2)=1
    const float p = pos[r < B ? r : r - B];
    acc += 2.0f + __logf(s) - p;
  }
  red[threadIdx.x] = acc;
  __syncthreads();
  for (int off = 128; off >= 1; off >>= 1) {
    if (threadIdx.x < off) red[threadIdx.x] += red[threadIdx.x + off];
    __syncthreads();
  }
  if (threadIdx.x == 0) out[0] = red[0] / (float)N;
}

extern "C" void kernel_launch(void* const* d_in, const int* in_sizes, int n_in,
                              void* d_out, int out_size, void* d_ws, size_t ws_size,
                              hipStream_t stream) {
  const float* z1 = (const float*)d_in[0];
  const float* z2 = (const float*)d_in[1];
  const int B = in_sizes[0] / D_DIM;  // 4096
  const int N = 2 * B;                // 8192

  float* zs      = (float*)d_ws;                 // [N][128] scaled normalized rows
  float* pos     = zs + (size_t)N * D_DIM;       // [B]
  float* partial = pos + B;                      // [COL_SPLIT][N] exp-sums
  float* out     = (float*)d_out;

  simclr_norm<<<(B + 7) / 8, 256, 0, stream>>>(z1, z2, zs, pos, B);

  const int colTiles  = N / (16 * COL_SPLIT);    // 64 column tiles per chunk
  const int rowGroups = N / (16 * 8);            // 64 groups of 8 row tiles
  simclr_sim<<<rowGroups * COL_SPLIT, 256, 0, stream>>>(zs, partial, N, colTiles);

  simclr_final<<<1, 256, 0, stream>>>(partial, pos, out, N, B);
}